// CrossAttention_33586644254982
// MI455X (gfx1250) — compile-verified
//
#include <hip/hip_runtime.h>
#include <hip/hip_bf16.h>

// ============================================================================
// CrossAttention with a SINGLE KV token: softmax over k=1 is identically 1,
// so the output is  broadcast_N( (freq @ Wv^T + bv) @ Wo^T + bo ).
// Q/Wq/bq/Wk/bk are mathematically dead.  Cost is dominated by the 201 MB
// output store (~8.6 us at 23.3 TB/s).  The two 16-row GEMMs use the f32
// WMMA path (V_WMMA_F32_16X16X4_F32) to keep full f32 precision.
// ============================================================================

typedef __attribute__((ext_vector_type(2))) float v2f;
typedef __attribute__((ext_vector_type(4))) float v4f;
typedef __attribute__((ext_vector_type(8))) float v8f;

// out[16, Ntot] = A[16, K] @ W[Ntot, K]^T + bias[Ntot]
// One wave (32 lanes) per 16-wide N tile.  EXEC is all-ones (blockDim=32),
// as WMMA requires.
__global__ __launch_bounds__(32) void gemm16_wmma_f32(
    const float* __restrict__ A,     // [16, K] row-major
    const float* __restrict__ W,     // [Ntot, K] row-major (acts as B = W^T)
    const float* __restrict__ bias,  // [Ntot]
    float* __restrict__ out,         // [16, Ntot]
    int K, int Ntot)
{
    const int lane  = threadIdx.x;        // 0..31
    const int n0    = blockIdx.x * 16;    // N tile origin
    const int row16 = lane & 15;          // M (for A) / N (for B) within tile
    const int khalf = (lane >> 4) * 2;    // lanes 16-31 hold the upper K pair

    // A lane L: (M = L%16, K = k0 + 2*(L/16) + {0,1})  -> contiguous pair
    const float* Arow = A + (size_t)row16 * K + khalf;
    // B lane L: (K = k0 + 2*(L/16) + {0,1}, N = n0 + L%16); B[k][n] = W[n*K+k]
    const float* Wrow = W + (size_t)(n0 + row16) * K + khalf;

    v8f acc = {};
    #pragma unroll 4
    for (int k0 = 0; k0 < K; k0 += 4) {
        v2f a = *(const v2f*)(Arow + k0);
        v2f b = *(const v2f*)(Wrow + k0);
        // 8 args: (neg_a, A, neg_b, B, c_mod, C, reuse_a, reuse_b)
        acc = __builtin_amdgcn_wmma_f32_16x16x4_f32(
            false, a, false, b, (short)0, acc, false, false);
    }

    // D layout: VGPR r -> M = r + 8*(lane/16), N = n0 + lane%16
    const int col   = n0 + row16;
    const int mbase = (lane >> 4) * 8;
    const float bb  = bias[col];
    #pragma unroll
    for (int r = 0; r < 8; ++r) {
        out[(size_t)(mbase + r) * Ntot + col] = acc[r] + bb;
    }
}

// out[b, n, :] = y[b, :] for all n.  Pure HBM-store bandwidth.
// Grid: 16 batches * 256 blocks, each block covers 16 rows of N.
// 192 threads * v4f = one 768-float C row per iteration.
__global__ __launch_bounds__(192) void broadcast_rows(
    const v4f* __restrict__ y4,      // [16, 192]  (float4 view of [16,768])
    v4f* __restrict__ out4)          // [16*4096, 192]
{
    const int b  = blockIdx.x >> 8;          // 256 blocks per batch
    const int n0 = (blockIdx.x & 255) * 16;  // 16 N-rows per block

    const v4f v = y4[b * 192 + threadIdx.x];
    size_t base = ((size_t)b * 4096 + (size_t)n0) * 192 + threadIdx.x;

    #pragma unroll
    for (int i = 0; i < 16; ++i) {
        // Output (~201MB) exceeds usable L2 and is consumed by the host:
        // stream it with non-temporal stores.
        __builtin_nontemporal_store(v, &out4[base + (size_t)i * 192]);
    }
}

extern "C" void kernel_launch(void* const* d_in, const int* in_sizes, int n_in,
                              void* d_out, int out_size, void* d_ws, size_t ws_size,
                              hipStream_t stream) {
    // Input order: spatial_tokens, freq_token, Wq, bq, Wk, bk, Wv, bv, Wo, bo
    const float* freq = (const float*)d_in[1];  // [16, 512]
    const float* Wv   = (const float*)d_in[6];  // [768, 512]
    const float* bv   = (const float*)d_in[7];  // [768]
    const float* Wo   = (const float*)d_in[8];  // [768, 768]
    const float* bo   = (const float*)d_in[9];  // [768]

    constexpr int B = 16, C = 768, Cfd = 512, N = 4096;

    float* Vtmp = (float*)d_ws;        // [16, 768]
    float* y    = Vtmp + B * C;        // [16, 768]

    // V = freq @ Wv^T + bv            (M=16, N=768, K=512)
    gemm16_wmma_f32<<<C / 16, 32, 0, stream>>>(freq, Wv, bv, Vtmp, Cfd, C);
    // y = V @ Wo^T + bo               (M=16, N=768, K=768)
    gemm16_wmma_f32<<<C / 16, 32, 0, stream>>>(Vtmp, Wo, bo, y, C, C);
    // out[b, n, :] = y[b, :]          (201 MB streaming store)
    broadcast_rows<<<B * (N / 16), 192, 0, stream>>>((const v4f*)y, (v4f*)d_out);
    (void)in_sizes; (void)n_in; (void)out_size; (void)ws_size;
}